// dLDS_continuous_84963043049625
// MI455X (gfx1250) — compile-verified
//
#include <hip/hip_runtime.h>
#include <hip/hip_bf16.h>

typedef __attribute__((ext_vector_type(2))) float v2f;
typedef __attribute__((ext_vector_type(8))) float v8f;

// ---------------------------------------------------------------------------
// 3x3 matmul (row-major), fully unrolled FMA form
// ---------------------------------------------------------------------------
__device__ __forceinline__ void mm3(const float* __restrict__ A,
                                    const float* __restrict__ Bm,
                                    float* __restrict__ D) {
#pragma unroll
  for (int r = 0; r < 3; ++r) {
    const float a0 = A[3 * r + 0], a1 = A[3 * r + 1], a2 = A[3 * r + 2];
#pragma unroll
    for (int q = 0; q < 3; ++q) {
      D[3 * r + q] = fmaf(a0, Bm[q], fmaf(a1, Bm[3 + q], a2 * Bm[6 + q]));
    }
  }
}

// ---------------------------------------------------------------------------
// y = expm(T) @ x  for a 3x3 row-major T, via scaling-and-squaring with a
// 10-term Horner Taylor series (scaled ||T||_inf <= 0.5 -> series error
// ~1e-11, far below f32 eps).
// ---------------------------------------------------------------------------
__device__ __forceinline__ void expm3_apply(const float t[9],
                                            const float xv[3],
                                            float yv[3]) {
  // inf-norm (max abs row sum)
  const float r0 = fabsf(t[0]) + fabsf(t[1]) + fabsf(t[2]);
  const float r1 = fabsf(t[3]) + fabsf(t[4]) + fabsf(t[5]);
  const float r2 = fabsf(t[6]) + fabsf(t[7]) + fabsf(t[8]);
  const float nrm = fmaxf(r0, fmaxf(r1, r2));

  int s = 0;
  if (nrm > 0.5f) {
    s = (int)ceilf(log2f(nrm)) + 1;  // nrm / 2^s <= 0.5
    s = s < 1 ? 1 : (s > 24 ? 24 : s);
  }
  const float sc = ldexpf(1.0f, -s);

  float Aa[9];
#pragma unroll
  for (int i = 0; i < 9; ++i) Aa[i] = t[i] * sc;

  // Horner: E = I + A(I + A/2(I + A/3(... (I + A/10))))
  const int NT = 10;
  float P[9];
#pragma unroll
  for (int i = 0; i < 9; ++i) P[i] = Aa[i] * (1.0f / NT);
  P[0] += 1.0f; P[4] += 1.0f; P[8] += 1.0f;

#pragma unroll
  for (int k = NT - 1; k >= 1; --k) {
    float Tm[9];
    mm3(Aa, P, Tm);
    const float rk = 1.0f / (float)k;  // compile-time constant after unroll
#pragma unroll
    for (int i = 0; i < 9; ++i) P[i] = Tm[i] * rk;
    P[0] += 1.0f; P[4] += 1.0f; P[8] += 1.0f;
  }

  // undo scaling: square s times
  for (int q = 0; q < s; ++q) {
    float Tm[9];
    mm3(P, P, Tm);
#pragma unroll
    for (int i = 0; i < 9; ++i) P[i] = Tm[i];
  }

  yv[0] = fmaf(P[0], xv[0], fmaf(P[1], xv[1], P[2] * xv[2]));
  yv[1] = fmaf(P[3], xv[0], fmaf(P[4], xv[1], P[5] * xv[2]));
  yv[2] = fmaf(P[6], xv[0], fmaf(P[7], xv[1], P[8] * xv[2]));
}

// ---------------------------------------------------------------------------
// Main kernel: 32 batches per wave. T-formation via V_WMMA_F32_16X16X4_F32:
//   D(16x16) = A(16x8) x Bc(8x16)
//   A[i,m]   = G[m][i % 9]   (rows 8..15 duplicate entries {8,0,1,...,6})
//   Bc[m,j]  = c[b0 (+16) + j, m]  (m=6,7 padded zero)
//   => D[i,j] = T[batch j] entry (i % 9)
// Each lane ends holding 8/9 entries of its own batch's T; the 9th comes from
// the partner lane (lane ^ 16) via one ds_swizzle SWAPX16. batch = b0 + lane,
// so x/out accesses are perfectly coalesced.
// All loads are unconditional (addresses valid for every lane); zero-padding
// of the K=6,7 columns is done arithmetically so no EXEC-divergent regions
// exist anywhere near the WMMAs.
// ---------------------------------------------------------------------------
__global__ __launch_bounds__(256) void dlds_expm_wmma_kernel(
    const float* __restrict__ x, const float* __restrict__ c,
    const float* __restrict__ G, float* __restrict__ out) {
  const int lane = threadIdx.x & 31;
  const int wave = threadIdx.x >> 5;
  const int j    = lane & 15;
  const int half = lane >> 4;
  const int bbase = blockIdx.x * 256 + wave * 32;
  const int myb   = bbase + lane;

  const float msk = half ? 0.0f : 1.0f;  // zero-pad mask for K=6,7 rows

  // ---- A operand (uniform across waves; per-lane row j, K = 2*half + {0,1})
  const int e  = j - (j >= 9 ? 9 : 0);   // j % 9
  const int m0 = 2 * half;
  v2f a0, a1;
  a0.x = G[m0 * 9 + e];
  a0.y = G[(m0 + 1) * 9 + e];
  a1.x = G[4 * 9 + e] * msk;             // K=4..5 live in half 0; half 1 is pad
  a1.y = G[5 * 9 + e] * msk;

  // ---- B operands: column j = batch, row m = coefficient index
  const int bA = bbase + j;
  const int bB = bbase + 16 + j;
  // byte addr = 24*b + 8*half (+16): always 8B aligned -> float2 loads.
  // All four loads are unconditional; masking is arithmetic.
  const float2 cA0 = *(const float2*)(c + 6 * bA + m0);
  const float2 cB0 = *(const float2*)(c + 6 * bB + m0);
  const float2 cA1 = *(const float2*)(c + 6 * bA + 4);
  const float2 cB1 = *(const float2*)(c + 6 * bB + 4);

  const v2f bA0 = {cA0.x, cA0.y}, bA1 = {cA1.x * msk, cA1.y * msk};
  const v2f bB0 = {cB0.x, cB0.y}, bB1 = {cB1.x * msk, cB1.y * msk};

  // ---- 4x V_WMMA_F32_16X16X4_F32 (exact f32, K = 0..3 then 4..7)
  v8f Da = {0.f, 0.f, 0.f, 0.f, 0.f, 0.f, 0.f, 0.f};
  v8f Db = {0.f, 0.f, 0.f, 0.f, 0.f, 0.f, 0.f, 0.f};
  Da = __builtin_amdgcn_wmma_f32_16x16x4_f32(false, a0, false, bA0, (short)0, Da, false, false);
  Da = __builtin_amdgcn_wmma_f32_16x16x4_f32(false, a1, false, bA1, (short)0, Da, false, false);
  Db = __builtin_amdgcn_wmma_f32_16x16x4_f32(false, a0, false, bB0, (short)0, Db, false, false);
  Db = __builtin_amdgcn_wmma_f32_16x16x4_f32(false, a1, false, bB1, (short)0, Db, false, false);

  // ---- exchange the one entry held by the partner lane (lane ^ 16)
  // lane j (half 0) needs D_a row 8 (entry 8) -> lives in lane 16+j, VGPR 0
  // lane 16+j (half 1) needs D_b row 7 (entry 7) -> lives in lane j, VGPR 7
  // ds_swizzle group-of-32: xor_mask=0x10, or=0, and=0x1f  => SWAPX16
  const float sendv = half ? Da[0] : Db[7];
  const float recv  = __uint_as_float(
      (unsigned)__builtin_amdgcn_ds_swizzle(__float_as_uint(sendv), 0x401F));

  float t[9];
  if (half == 0) {  // rows 0..7 of D_a are entries 0..7 of batch bbase+j
#pragma unroll
    for (int i = 0; i < 8; ++i) t[i] = Da[i];
    t[8] = recv;
  } else {          // rows 8..15 of D_b are entries {8,0,1,2,3,4,5,6}
    t[8] = Db[0];
#pragma unroll
    for (int i = 0; i < 7; ++i) t[i] = Db[i + 1];
    t[7] = recv;
  }

  // ---- per-lane expm + matvec (coalesced x/out: batch = bbase + lane)
  const float xv[3] = {x[3 * myb + 0], x[3 * myb + 1], x[3 * myb + 2]};
  float yv[3];
  expm3_apply(t, xv, yv);
  out[3 * myb + 0] = yv[0];
  out[3 * myb + 1] = yv[1];
  out[3 * myb + 2] = yv[2];
}

// ---------------------------------------------------------------------------
// Scalar tail (only used if B % 256 != 0; not the case for B = 2^21)
// ---------------------------------------------------------------------------
__global__ __launch_bounds__(256) void dlds_expm_tail_kernel(
    const float* __restrict__ x, const float* __restrict__ c,
    const float* __restrict__ G, float* __restrict__ out, int bstart, int btotal) {
  const int b = bstart + blockIdx.x * blockDim.x + threadIdx.x;
  if (b >= btotal) return;
  float t[9];
#pragma unroll
  for (int i = 0; i < 9; ++i) {
    float acc = 0.0f;
#pragma unroll
    for (int m = 0; m < 6; ++m) acc = fmaf(c[6 * b + m], G[9 * m + i], acc);
    t[i] = acc;
  }
  const float xv[3] = {x[3 * b + 0], x[3 * b + 1], x[3 * b + 2]};
  float yv[3];
  expm3_apply(t, xv, yv);
  out[3 * b + 0] = yv[0];
  out[3 * b + 1] = yv[1];
  out[3 * b + 2] = yv[2];
}

extern "C" void kernel_launch(void* const* d_in, const int* in_sizes, int n_in,
                              void* d_out, int out_size, void* d_ws, size_t ws_size,
                              hipStream_t stream) {
  const float* x = (const float*)d_in[0];  // (B, 3, 1) f32
  const float* c = (const float*)d_in[1];  // (B, 6)    f32
  const float* G = (const float*)d_in[2];  // (6, 3, 3) f32
  float* out = (float*)d_out;              // (B, 3, 1) f32

  const int B = in_sizes[0] / 3;
  const int Bmain = (B / 256) * 256;
  if (Bmain > 0) {
    dlds_expm_wmma_kernel<<<Bmain / 256, 256, 0, stream>>>(x, c, G, out);
  }
  const int rem = B - Bmain;
  if (rem > 0) {
    dlds_expm_tail_kernel<<<(rem + 255) / 256, 256, 0, stream>>>(x, c, G, out, Bmain, B);
  }
}